// HausdorffLoss_13872744366713
// MI455X (gfx1250) — compile-verified
//
#include <hip/hip_runtime.h>
#include <math.h>

typedef __attribute__((ext_vector_type(2))) float v2f;
typedef __attribute__((ext_vector_type(8))) float v8f;

#define HD_B 4
#define HD_N 8192
#define HD_M 8192
#define HD_TILES (HD_M / 16)   // 512 column tiles (compile-time, divisible by 4)

// Fused directed-min kernel, both directions via blockIdx.z.
//   z==0: A=pred, B=targ -> out[0*B*N + b*N + i] = min_j d2(pred_i, targ_j)
//   z==1: A=targ, B=pred -> out[1*B*N + b*N + j] = min_i d2(pred_i, targ_j)
// grid = (N/128, B, 2), block = 256 (8 waves, each wave owns a 16-row tile).
__global__ __launch_bounds__(256)
void hausdorff_rowmin_kernel(const float* __restrict__ pred,
                             const float* __restrict__ targ,
                             float* __restrict__ outmin2) {
    // Pre-built per-lane WMMA B fragments, one 32-lane record per column tile:
    //   slots 0-15 : (x_j, y_j)         (K0,K1 rows of augmented B)
    //   slots 16-31: (1.0, |b_j|^2)     (K2,K3 rows of augmented B)
    __shared__ float2 fragLDS[HD_TILES * 32];   // 128 KB

    const int dir  = blockIdx.z;
    const int b    = blockIdx.y;
    const int tid  = threadIdx.x;
    const int lane = tid & 31;
    const int wave = tid >> 5;

    const float* __restrict__ Ab =
        (dir == 0 ? pred : targ) + (size_t)b * HD_N * 2;
    const float* __restrict__ Bb =
        (dir == 0 ? targ : pred) + (size_t)b * HD_M * 2;
    float* __restrict__ dst = outmin2 + (size_t)dir * HD_B * HD_N;

    // Precompute all column-tile fragments (coalesced b64 global loads).
    for (int j = tid; j < HD_M; j += 256) {
        const float bx = Bb[2 * j];
        const float by = Bb[2 * j + 1];
        const int t = j >> 4, s = j & 15;
        fragLDS[t * 32 + s]      = make_float2(bx, by);
        fragLDS[t * 32 + 16 + s] = make_float2(1.0f, bx * bx + by * by);
    }
    __syncthreads();

    // A fragment for this wave's 16-row tile (ISA 16x4 f32 A layout):
    //   lanes 0-15 : K0,K1 -> (-2x, -2y)
    //   lanes 16-31: K2,K3 -> (|a|^2, 1)
    const int rowBase = blockIdx.x * 128 + wave * 16;
    const int r = rowBase + (lane & 15);
    const float ax = Ab[2 * r];
    const float ay = Ab[2 * r + 1];
    v2f afrag;
    if (lane < 16) { afrag.x = -2.0f * ax; afrag.y = -2.0f * ay; }
    else           { afrag.x = ax * ax + ay * ay; afrag.y = 1.0f; }

    v8f rmin;
#pragma unroll
    for (int v = 0; v < 8; ++v) rmin[v] = 3.4e38f;

    const v2f* __restrict__ fragv = (const v2f*)fragLDS;
#pragma unroll 4
    for (int ct = 0; ct < HD_TILES; ++ct) {
        v2f bfrag = fragv[ct * 32 + lane];      // ds_load_2addr_b64 pairs
        v8f czero = {};
        // D = A x B + 0 == full 16x16 tile of squared distances in one WMMA.
        v8f d = __builtin_amdgcn_wmma_f32_16x16x4_f32(
            false, afrag, false, bfrag, (short)0, czero, false, false);
#pragma unroll
        for (int v = 0; v < 8; ++v) rmin[v] = fminf(rmin[v], d[v]);
    }

    // Min across the 16 lanes of each half (tile columns).
#pragma unroll
    for (int v = 0; v < 8; ++v) {
        float m = rmin[v];
#pragma unroll
        for (int off = 1; off < 16; off <<= 1)
            m = fminf(m, __shfl_xor(m, off, 32));
        rmin[v] = m;
    }

    // C layout: element (row = v + 8*(lane>=16), col = lane&15). Lanes 0 and
    // 16 hold the reduced mins for rows rowBase+0..7 / rowBase+8..15.
    if ((lane & 15) == 0) {
        const int base = rowBase + (lane >> 4) * 8;
#pragma unroll
        for (int v = 0; v < 8; ++v)
            dst[(size_t)b * HD_N + base + v] = rmin[v];
    }
}

// Final: per-batch max over both directed minima, sqrt, mean over batches.
__global__ __launch_bounds__(256)
void hausdorff_final_kernel(const float* __restrict__ rowmin2,
                            const float* __restrict__ colmin2,
                            float* __restrict__ out) {
    __shared__ float red[256];
    float sum = 0.0f;
    for (int b = 0; b < HD_B; ++b) {
        float mx = 0.0f;
        for (int i = threadIdx.x; i < HD_N; i += 256)
            mx = fmaxf(mx, rowmin2[(size_t)b * HD_N + i]);
        for (int j = threadIdx.x; j < HD_M; j += 256)
            mx = fmaxf(mx, colmin2[(size_t)b * HD_M + j]);
        red[threadIdx.x] = mx;
        __syncthreads();
        for (int s = 128; s > 0; s >>= 1) {
            if (threadIdx.x < s)
                red[threadIdx.x] = fmaxf(red[threadIdx.x], red[threadIdx.x + s]);
            __syncthreads();
        }
        if (threadIdx.x == 0) sum += sqrtf(fmaxf(red[0], 0.0f));
        __syncthreads();
    }
    if (threadIdx.x == 0) out[0] = sum / (float)HD_B;
}

extern "C" void kernel_launch(void* const* d_in, const int* in_sizes, int n_in,
                              void* d_out, int out_size, void* d_ws, size_t ws_size,
                              hipStream_t stream) {
    const float* pred = (const float*)d_in[0];  // [B, N, 2]
    const float* targ = (const float*)d_in[1];  // [B, M, 2]
    float* out = (float*)d_out;

    float* rowmin2 = (float*)d_ws;                     // [2][B*N]: dir0, dir1
    float* colmin2 = rowmin2 + (size_t)HD_B * HD_N;

    dim3 block(256);
    hausdorff_rowmin_kernel<<<dim3(HD_N / 128, HD_B, 2), block, 0, stream>>>(
        pred, targ, rowmin2);
    hausdorff_final_kernel<<<1, block, 0, stream>>>(rowmin2, colmin2, out);
}